// DecodePredictions_33870112096818
// MI455X (gfx1250) — compile-verified
//
#include <hip/hip_runtime.h>
#include <cfloat>

// ---------------------------------------------------------------------------
// DecodePredictions for gfx1250 (MI455X)
//   predictions [1, N, 4+80] f32, anchors [N,4] f32 ->
//   det_boxes [100,4], det_ids [100], det_probs [100]  (600 floats, concat)
// ---------------------------------------------------------------------------

#define NANCH   306900
#define ROWF4   21          // 84 floats = 21 float4 per anchor row
#define TOPK    1000
#define MAXDET  100
#define SORTN   4096
#define NBINS   2048
#define IMGSZ   1280.0f
#define NEG_INF (-__builtin_inff())

typedef __attribute__((ext_vector_type(4))) unsigned v4u;
typedef __attribute__((ext_vector_type(4))) int      v4i;
typedef __attribute__((ext_vector_type(8))) int      v8i;

// Workspace layout in 4-byte units (total ~7.1 MB)
#define SC_OFF   0                       // float[N]   scores
#define ID_OFF   (NANCH)                 // int[N]     class ids
#define BX_OFF   (2*NANCH)               // float[4N]  decoded corner boxes
#define HIST_OFF (6*NANCH)               // u32[2048]
#define CNT_OFF  (6*NANCH + NBINS)       // u32[4]
#define THR_OFF  (CNT_OFF + 4)           // u32[4]
#define KEY_OFF  (THR_OFF + 4)           // u64[SORTN]  (byte offset % 8 == 0)
#define TS_OFF   (KEY_OFF + 2*SORTN)     // float[1024] top-k scores
#define TB_OFF   (TS_OFF + 1024)         // float[4096] top-k boxes (16B aligned)
#define TI_OFF   (TB_OFF + 4096)         // int[1024]   top-k ids

// ------------------------------- kernels -----------------------------------

__global__ __launch_bounds__(256) void k_zero(unsigned* __restrict__ meta) {
  for (int i = threadIdx.x; i < NBINS + 8; i += 256) meta[i] = 0u;
}

__global__ __launch_bounds__(256) void k_decode(const float* __restrict__ pred,
                                                const float* __restrict__ anchors,
                                                float* __restrict__ ws) {
  int i = blockIdx.x * 256 + threadIdx.x;
  if (i >= NANCH) return;
  const float4* row = reinterpret_cast<const float4*>(pred) + (size_t)i * ROWF4;
  // gfx1250 global_prefetch_b8: pull the stream ~256 rows ahead into cache.
  __builtin_prefetch(reinterpret_cast<const char*>(row) + 256 * 336, 0, 0);

  float4 d = row[0];                       // dx,dy,dw,dh
  float m = -FLT_MAX; int arg = 0;         // argmax over 80 logits (first max wins)
#pragma unroll
  for (int q = 1; q <= 20; ++q) {
    float4 v = row[q];
    int base = q * 4 - 4;
    if (v.x > m) { m = v.x; arg = base; }
    if (v.y > m) { m = v.y; arg = base + 1; }
    if (v.z > m) { m = v.z; arg = base + 2; }
    if (v.w > m) { m = v.w; arg = base + 3; }
  }
  float prob = 1.0f / (1.0f + __expf(-m)); // sigmoid monotone => max(sigmoid)=sigmoid(max)

  float4 a = reinterpret_cast<const float4*>(anchors)[i]; // cx,cy,w,h
  float cx = d.x * 0.1f * a.z + a.x;
  float cy = d.y * 0.1f * a.w + a.y;
  float hw = 0.5f * __expf(d.z * 0.2f) * a.z;
  float hh = 0.5f * __expf(d.w * 0.2f) * a.w;
  float4 c;
  c.x = fminf(fmaxf(cx - hw, 0.0f), IMGSZ);
  c.y = fminf(fmaxf(cy - hh, 0.0f), IMGSZ);
  c.z = fminf(fmaxf(cx + hw, 0.0f), IMGSZ);
  c.w = fminf(fmaxf(cy + hh, 0.0f), IMGSZ);

  ws[SC_OFF + i] = prob;
  reinterpret_cast<int*>(ws)[ID_OFF + i] = arg;
  reinterpret_cast<float4*>(ws + BX_OFF)[i] = c;
}

__device__ __forceinline__ unsigned score_bin(float s) {
  unsigned b = __float_as_uint(s) >> 19;   // probs in (0,1): bits monotone, bins 0..2031
  return b >= NBINS ? NBINS - 1 : b;
}

__global__ __launch_bounds__(256) void k_hist(const float* __restrict__ sc,
                                              unsigned* __restrict__ hist) {
  __shared__ unsigned lh[NBINS];
  for (int b = threadIdx.x; b < NBINS; b += 256) lh[b] = 0u;
  __syncthreads();
  int i = blockIdx.x * 256 + threadIdx.x;
  if (i < NANCH) atomicAdd(&lh[score_bin(sc[i])], 1u);
  __syncthreads();
  for (int b = threadIdx.x; b < NBINS; b += 256)
    if (lh[b]) atomicAdd(&hist[b], lh[b]);
}

__global__ void k_thresh(const unsigned* __restrict__ hist, unsigned* __restrict__ thr) {
  if (threadIdx.x == 0) {
    unsigned cum = 0; int b = NBINS - 1;
    for (; b >= 0; --b) { cum += hist[b]; if (cum >= TOPK) break; }
    thr[0] = (b < 0) ? 0u : (unsigned)b;
  }
}

__global__ __launch_bounds__(256) void k_compact(const float* __restrict__ sc,
                                                 const unsigned* __restrict__ thr,
                                                 unsigned long long* __restrict__ keys,
                                                 unsigned* __restrict__ counter) {
  int i = blockIdx.x * 256 + threadIdx.x;
  if (i >= NANCH) return;
  unsigned bits = __float_as_uint(sc[i]);
  unsigned bin = bits >> 19; if (bin >= NBINS) bin = NBINS - 1;
  if (bin >= thr[0]) {
    unsigned pos = atomicAdd(counter, 1u);
    if (pos < SORTN)  // key = (score bits, ~index): total order => deterministic sort
      keys[pos] = ((unsigned long long)bits << 32) | (unsigned)(~(unsigned)i);
  }
}

__global__ __launch_bounds__(1024) void k_sortgather(const float* __restrict__ ws,
                                                     const unsigned long long* __restrict__ gkeys,
                                                     const unsigned* __restrict__ counter,
                                                     float* __restrict__ ts,
                                                     float* __restrict__ tb,
                                                     int* __restrict__ ti) {
  __shared__ unsigned long long sk[SORTN];   // 32 KB LDS
  unsigned cnt = counter[0]; if (cnt > SORTN) cnt = SORTN;
  for (unsigned i = threadIdx.x; i < SORTN; i += 1024)
    sk[i] = (i < cnt) ? gkeys[i] : 0ULL;
  // Bitonic sort, descending.
  for (unsigned k = 2; k <= SORTN; k <<= 1)
    for (unsigned j = k >> 1; j > 0; j >>= 1) {
      __syncthreads();
      for (unsigned i = threadIdx.x; i < SORTN; i += 1024) {
        unsigned p = i ^ j;
        if (p > i) {
          unsigned long long a = sk[i], b = sk[p];
          bool up = ((i & k) == 0);
          if (up ? (a < b) : (a > b)) { sk[i] = b; sk[p] = a; }
        }
      }
    }
  __syncthreads();
  unsigned s = threadIdx.x;
  if (s < TOPK) {
    if (s < cnt) {
      unsigned long long key = sk[s];
      unsigned idx = ~(unsigned)(key & 0xFFFFFFFFu);
      ts[s] = __uint_as_float((unsigned)(key >> 32));
      ti[s] = reinterpret_cast<const int*>(ws)[ID_OFF + idx];
      reinterpret_cast<float4*>(tb)[s] = reinterpret_cast<const float4*>(ws + BX_OFF)[idx];
    } else {
      ts[s] = NEG_INF; ti[s] = -1;
      reinterpret_cast<float4*>(tb)[s] = make_float4(0.f, 0.f, 0.f, 0.f);
    }
  }
}

struct __align__(16) NmsSmem {
  float box[TOPK][4];   // MUST be at LDS offset 0: TDM lds_addr = 0 targets this
  float sc[1024];
  int   id[1024];
  float rv[32];
  int   ri[32];
  float bestv;
  int   besti;
};

__global__ __launch_bounds__(1024) void k_nms(const float* __restrict__ ts,
                                              const float* __restrict__ tb,
                                              const int* __restrict__ ti,
                                              float* __restrict__ out) {
  __shared__ NmsSmem sm;
  int tid = threadIdx.x;

#if __has_builtin(__builtin_amdgcn_tensor_load_to_lds)
  // Tensor Data Mover: DMA the 1000x4 f32 box tile (16000 B) into LDS offset 0.
  if (tid < 32) {  // one wave issues; TDM is a wave-level op, EXEC ignored
    unsigned long long ga = (unsigned long long)(size_t)tb;
    v4u g0;
    g0[0] = 1u;                                                  // count=1 valid D#
    g0[1] = 0u;                                                  // lds_addr = 0
    g0[2] = (unsigned)(ga & 0xFFFFFFFFu);                        // global_addr lo
    g0[3] = ((unsigned)(ga >> 32) & 0x01FFFFFFu) | 0x80000000u;  // addr hi | type=2
    // data_size=4B; tensor_dim0=4000, tensor_dim1=1; tile_dim0=4000, tile_dim1=1;
    // tensor_dim0_stride=4000; no pad/iterate/multicast.
    v8i g1 = {0x00020000, 0x0FA00000, 0x00010000, 0x0FA00000, 1, 4000, 0, 0};
    v4i g2 = {0, 0, 0, 0};
    v4i g3 = {0, 0, 0, 0};
#if __clang_major__ >= 23
    v8i g4 = {0, 0, 0, 0, 0, 0, 0, 0};
    __builtin_amdgcn_tensor_load_to_lds(g0, g1, g2, g3, g4, 0);
#else
    __builtin_amdgcn_tensor_load_to_lds(g0, g1, g2, g3, 0);
#endif
#if __has_builtin(__builtin_amdgcn_s_wait_tensorcnt)
    __builtin_amdgcn_s_wait_tensorcnt((short)0);
#endif
  }
#else
  if (tid < TOPK) {
    float4 b = reinterpret_cast<const float4*>(tb)[tid];
    sm.box[tid][0] = b.x; sm.box[tid][1] = b.y;
    sm.box[tid][2] = b.z; sm.box[tid][3] = b.w;
  }
#endif

  if (tid < TOPK) {
    float s = ts[tid];
    sm.sc[tid] = (s > 0.5f) ? s : NEG_INF;   // CONF_THR mask
    sm.id[tid] = ti[tid];
  } else {
    sm.sc[tid] = NEG_INF;
    sm.id[tid] = -1;
  }
  __syncthreads();

  for (int t = 0; t < MAXDET; ++t) {
    // Block argmax with first-occurrence tie-break (matches jnp.argmax).
    float v = sm.sc[tid];
    int   idx = tid;
#pragma unroll
    for (int off = 16; off > 0; off >>= 1) {
      float v2 = __shfl_down(v, off, 32);
      int   i2 = __shfl_down(idx, off, 32);
      if (v2 > v || (v2 == v && i2 < idx)) { v = v2; idx = i2; }
    }
    if ((tid & 31) == 0) { sm.rv[tid >> 5] = v; sm.ri[tid >> 5] = idx; }
    __syncthreads();
    if (tid < 32) {
      v = sm.rv[tid]; idx = sm.ri[tid];
#pragma unroll
      for (int off = 16; off > 0; off >>= 1) {
        float v2 = __shfl_down(v, off, 32);
        int   i2 = __shfl_down(idx, off, 32);
        if (v2 > v || (v2 == v && i2 < idx)) { v = v2; idx = i2; }
      }
      if (tid == 0) { sm.bestv = v; sm.besti = idx; }
    }
    __syncthreads();

    float bv = sm.bestv; int bi = sm.besti;
    bool valid = bv > NEG_INF;
    float bx1 = sm.box[bi][0], by1 = sm.box[bi][1];
    float bx2 = sm.box[bi][2], by2 = sm.box[bi][3];

    if (valid) {
      if (tid < TOPK) {
        float x1 = fmaxf(bx1, sm.box[tid][0]);
        float y1 = fmaxf(by1, sm.box[tid][1]);
        float x2 = fminf(bx2, sm.box[tid][2]);
        float y2 = fminf(by2, sm.box[tid][3]);
        float iw = fmaxf(x2 - x1, 0.0f), ih = fmaxf(y2 - y1, 0.0f);
        float inter = iw * ih;
        float a1 = (bx2 - bx1) * (by2 - by1);
        float a2 = (sm.box[tid][2] - sm.box[tid][0]) * (sm.box[tid][3] - sm.box[tid][1]);
        float iou = inter / (a1 + a2 - inter + 1e-8f);
        if (iou > 0.5f) sm.sc[tid] = NEG_INF;
      }
      if (tid == 0) sm.sc[bi] = NEG_INF;   // explicit .at[i].set(-inf)
    }
    if (tid == 0) {
      if (valid) {
        out[4 * t + 0] = bx1; out[4 * t + 1] = by1;
        out[4 * t + 2] = bx2; out[4 * t + 3] = by2;
        out[400 + t] = (float)sm.id[bi];
        out[500 + t] = bv;
      } else {
        out[4 * t + 0] = 0.f; out[4 * t + 1] = 0.f;
        out[4 * t + 2] = 0.f; out[4 * t + 3] = 0.f;
        out[400 + t] = -1.0f;
        out[500 + t] = 0.0f;
      }
    }
    __syncthreads();
  }
}

// ------------------------------ launcher -----------------------------------

extern "C" void kernel_launch(void* const* d_in, const int* in_sizes, int n_in,
                              void* d_out, int out_size, void* d_ws, size_t ws_size,
                              hipStream_t stream) {
  (void)in_sizes; (void)n_in; (void)out_size; (void)ws_size;
  const float* pred = (const float*)d_in[0];
  const float* anch = (const float*)d_in[1];
  float*    out = (float*)d_out;
  float*    ws  = (float*)d_ws;
  unsigned* wsu = (unsigned*)d_ws;
  unsigned long long* keys = (unsigned long long*)(wsu + KEY_OFF);

  const int nb = (NANCH + 255) / 256;
  k_zero<<<1, 256, 0, stream>>>(wsu + HIST_OFF);
  k_decode<<<nb, 256, 0, stream>>>(pred, anch, ws);
  k_hist<<<nb, 256, 0, stream>>>(ws + SC_OFF, wsu + HIST_OFF);
  k_thresh<<<1, 32, 0, stream>>>(wsu + HIST_OFF, wsu + THR_OFF);
  k_compact<<<nb, 256, 0, stream>>>(ws + SC_OFF, wsu + THR_OFF, keys, wsu + CNT_OFF);
  k_sortgather<<<1, 1024, 0, stream>>>(ws, keys, wsu + CNT_OFF,
                                       ws + TS_OFF, ws + TB_OFF, (int*)(wsu + TI_OFF));
  k_nms<<<1, 1024, 0, stream>>>(ws + TS_OFF, ws + TB_OFF,
                                (const int*)(wsu + TI_OFF), out);
}